// GradientSim_28226525070040
// MI455X (gfx1250) — compile-verified
//
#include <hip/hip_runtime.h>

// Problem geometry
#define IX 160
#define IY 192
#define IZ 160
#define GX 159   // gradient/output extents
#define GY 191
#define GZ 159
#define VOL ((size_t)GX * GY * GZ)   // per-batch per-component volume
#define EPSF 0.01f

typedef float v2f __attribute__((ext_vector_type(2)));
typedef float v8f __attribute__((ext_vector_type(8)));

// ---------------------------------------------------------------------------
// Kernel 0: zero the f64 accumulator (workspace is poisoned with 0xAA).
// ---------------------------------------------------------------------------
__global__ void k_init(double* acc) {
  if (threadIdx.x == 0 && blockIdx.x == 0) *acc = 0.0;
}

// ---------------------------------------------------------------------------
// Kernel 1: per (b,x,y) line along z — compute dx,dy,dz of both images on the
// fly and the 7-tap centered z box-sum; write 6 component volumes to scratch.
// Coalesced along z; scratch is L2-resident (116 MB < 192 MB L2).
// ---------------------------------------------------------------------------
__global__ void __launch_bounds__(IZ)
k_grad_zsum(const float* __restrict__ Ii, const float* __restrict__ Ji,
            float* __restrict__ T1, int b) {
  const int line = blockIdx.x;        // 0 .. GX*GY-1
  const int x = line / GY;            // 0..158
  const int y = line % GY;            // 0..190
  const int t = threadIdx.x;          // 0..159 (z)

  __shared__ float g[6][IZ];          // dxI,dyI,dzI,dxJ,dyJ,dzJ at z-index t

  const float* imgs[2] = {Ii, Ji};
  #pragma unroll
  for (int im = 0; im < 2; ++im) {
    const float* I = imgs[im];
    const size_t o11 = (((size_t)b * IX + (x + 1)) * IY + (y + 1)) * IZ;
    const size_t o01 = (((size_t)b * IX +  x     ) * IY + (y + 1)) * IZ;
    const size_t o10 = (((size_t)b * IX + (x + 1)) * IY +  y     ) * IZ;
    const float a_t  = I[o11 + t];                                // I[x+1][y+1][t]
    const float a_t1 = (t + 1 < IZ) ? I[o11 + t + 1] : 0.f;       // I[x+1][y+1][t+1]
    const float b01  = (t + 1 < IZ) ? I[o01 + t + 1] : 0.f;       // I[x  ][y+1][t+1]
    const float c10  = (t + 1 < IZ) ? I[o10 + t + 1] : 0.f;       // I[x+1][y  ][t+1]
    if (t < GZ) {
      g[im * 3 + 0][t] = a_t1 - b01;   // dx
      g[im * 3 + 1][t] = a_t1 - c10;   // dy
      g[im * 3 + 2][t] = a_t1 - a_t;   // dz
    }
  }
  __syncthreads();

  if (t < GZ) {
    const int lo = (t - 3 < 0) ? 0 : t - 3;
    const int hi = (t + 3 > GZ - 1) ? GZ - 1 : t + 3;
    float s[6] = {0.f, 0.f, 0.f, 0.f, 0.f, 0.f};
    for (int k = lo; k <= hi; ++k) {
      #pragma unroll
      for (int c = 0; c < 6; ++c) s[c] += g[c][k];
    }
    const size_t o = ((size_t)x * GY + y) * GZ + t;
    #pragma unroll
    for (int c = 0; c < 6; ++c) T1[(size_t)c * VOL + o] = s[c];
  }
}

// ---------------------------------------------------------------------------
// Kernel 2: one wave32 per 16x16 (x,y) tile at fixed z. Load 22x22 halo of
// all 6 z-summed components to LDS, then do the y-pass and x-pass 7-tap sums
// as banded GEMMs on the WMMA unit (band matrix is exact 0/1 in f32), then
// the cross/norm combine + reduction into a global f64 accumulator.
// ---------------------------------------------------------------------------
#define TX 10   // ceil(159/16)
#define TY 12   // ceil(191/16)

__global__ void __launch_bounds__(32)
k_box_xy_combine(const float* __restrict__ T1, double* __restrict__ acc) {
  const int tile = blockIdx.x;          // 0 .. TX*TY-1
  const int tx = tile / TY;
  const int ty = tile % TY;
  const int x0 = tx * 16;
  const int y0 = ty * 16;
  const int z  = blockIdx.y;            // 0..GZ-1
  const int lane = threadIdx.x;         // wave32

  __shared__ float din[6][32][24];      // halo tile per component (zero padded)
  __shared__ float P[32][16];           // y-summed intermediate (per component)
  __shared__ float red[32];

  // Fill LDS: rows ix in [0,22) -> x = x0-3+ix ; cols iy in [0,22) -> y = y0-3+iy.
  for (int idx = lane; idx < 6 * 32 * 24; idx += 32) {
    const int c  = idx / (32 * 24);
    const int r  = idx % (32 * 24);
    const int ix = r / 24;
    const int iy = r % 24;
    const int gx = x0 - 3 + ix;
    const int gy = y0 - 3 + iy;
    float v = 0.f;
    if (ix < 22 && iy < 22 && gx >= 0 && gx < GX && gy >= 0 && gy < GY)
      v = T1[(size_t)c * VOL + ((size_t)gx * GY + gy) * GZ + z];
    din[c][ix][iy] = v;
  }
  __syncthreads();

  const int n    = lane & 15;               // N index (also A's M index)
  const int koff = (lane < 16) ? 0 : 2;     // K base held by this lane-half

  v8f S[6];

  for (int c = 0; c < 6; ++c) {
    // ---- y-pass: P(ix, n) = sum_k din(ix, k) * Band[k][n], Band 1 iff k in [n, n+6]
    #pragma unroll
    for (int half = 0; half < 2; ++half) {
      v8f p = {};
      #pragma unroll
      for (int t = 0; t < 6; ++t) {
        const int kk = 4 * t + koff;
        v2f a, bb;
        a.x  = din[c][half * 16 + n][kk];
        a.y  = din[c][half * 16 + n][kk + 1];
        bb.x = (kk     >= n && kk     <= n + 6) ? 1.f : 0.f;
        bb.y = (kk + 1 >= n && kk + 1 <= n + 6) ? 1.f : 0.f;
        p = __builtin_amdgcn_wmma_f32_16x16x4_f32(false, a, false, bb,
                                                  (short)0, p, false, false);
      }
      const int mbase = ((lane < 16) ? 0 : 8) + half * 16;
      #pragma unroll
      for (int r = 0; r < 8; ++r) P[mbase + r][n] = p[r];
    }
    __syncthreads();

    // ---- x-pass: S(m, n) = sum_k Band[m][k] * P(k, n), Band 1 iff k in [m, m+6]
    v8f s = {};
    #pragma unroll
    for (int t = 0; t < 6; ++t) {
      const int kk = 4 * t + koff;
      v2f a, bb;
      a.x  = (kk     >= n && kk     <= n + 6) ? 1.f : 0.f;  // A row m == lane&15
      a.y  = (kk + 1 >= n && kk + 1 <= n + 6) ? 1.f : 0.f;
      bb.x = P[kk][n];
      bb.y = P[kk + 1][n];
      s = __builtin_amdgcn_wmma_f32_16x16x4_f32(false, a, false, bb,
                                                (short)0, s, false, false);
    }
    __syncthreads();   // before next component reuses P
    S[c] = s;
  }

  // ---- combine + reduce. Lane holds (m = r + [lane>=16]*8, n) of the 16x16 tile.
  float partial = 0.f;
  const int mb = (lane < 16) ? 0 : 8;
  #pragma unroll
  for (int r = 0; r < 8; ++r) {
    const int gx = x0 + mb + r;
    const int gy = y0 + n;
    if (gx < GX && gy < GY) {
      const float sxI = S[0][r], syI = S[1][r], szI = S[2][r];
      const float sxJ = S[3][r], syJ = S[4][r], szJ = S[5][r];
      const float cross = fabsf(sxI * sxJ + syI * syJ + szI * szJ) + EPSF;
      const float dI = sxI * sxI + syI * syI + szI * szI + EPSF;
      const float dJ = sxJ * sxJ + syJ * syJ + szJ * szJ + EPSF;
      partial += cross * rsqrtf(dI * dJ);
    }
  }
  red[lane] = partial;
  __syncthreads();
  if (lane == 0) {
    float tot = 0.f;
    #pragma unroll
    for (int i = 0; i < 32; ++i) tot += red[i];
    atomicAdd(acc, (double)tot);
  }
}

// ---------------------------------------------------------------------------
// Kernel 3: finalize mean into d_out[0].
// ---------------------------------------------------------------------------
__global__ void k_final(const double* acc, float* out) {
  if (threadIdx.x == 0 && blockIdx.x == 0)
    out[0] = (float)(*acc / (double)(2.0 * GX * GY * GZ));
}

// ---------------------------------------------------------------------------
extern "C" void kernel_launch(void* const* d_in, const int* in_sizes, int n_in,
                              void* d_out, int out_size, void* d_ws, size_t ws_size,
                              hipStream_t stream) {
  (void)in_sizes; (void)n_in; (void)out_size; (void)ws_size;
  const float* Ii = (const float*)d_in[0];
  const float* Ji = (const float*)d_in[1];
  float* out = (float*)d_out;

  double* acc = (double*)d_ws;                       // 8 B accumulator
  float*  T1  = (float*)((char*)d_ws + 256);         // 6 * VOL f32 (~110.5 MiB)

  hipLaunchKernelGGL(k_init, dim3(1), dim3(32), 0, stream, acc);
  for (int b = 0; b < 2; ++b) {
    hipLaunchKernelGGL(k_grad_zsum, dim3(GX * GY), dim3(IZ), 0, stream,
                       Ii, Ji, T1, b);
    hipLaunchKernelGGL(k_box_xy_combine, dim3(TX * TY, GZ), dim3(32), 0, stream,
                       T1, acc);
  }
  hipLaunchKernelGGL(k_final, dim3(1), dim3(1), 0, stream, acc, out);
}